// GAT_11948599017832
// MI455X (gfx1250) — compile-verified
//
#include <hip/hip_runtime.h>
#include <math.h>

#define Bdim 4
#define Ndim 128
#define INdim 128
#define OUTdim 128
#define Hdim 8
#define Tdim 8
#define ZD 256
#define EC 32
#define EPSc 1e-5f

typedef float v2f __attribute__((ext_vector_type(2)));
typedef float v8f __attribute__((ext_vector_type(8)));

// workspace layout (float offsets)
#define WS_VALUES 0         // 65536  (B,N,OUT)
#define WS_SKIP   65536     // 65536  (B,N,OUT)
#define WS_A1     131072    // 4096   (B,N,H)
#define WS_A2     135168    // 4096
#define WS_TRI1   139264    // 4096   (B,N,T)
#define WS_TRI2   143360    // 4096
#define WS_AG     147456    // 32     (B,H)
#define WS_G      147488    // 32     (B,T)
#define WS_WCAT   147520    // 4096   (128 x 32): [ae | te1 | te2 | te3]
#define WS_EDGE   151616    // 2097152 (B*N*N, 32): [ae | e1+b | e2+b | e3+b]

// ---------------------------------------------------------------- K0: pack W
__global__ void build_wcat_kernel(const float* __restrict__ ae_w,
                                  const float* __restrict__ te1_w,
                                  const float* __restrict__ te2_w,
                                  const float* __restrict__ te3_w,
                                  float* __restrict__ ws) {
  int e = blockIdx.x * blockDim.x + threadIdx.x;   // 0..4095
  if (e >= INdim * EC) return;
  int k = e >> 5, c = e & 31;
  float v;
  if (c < 8)       v = ae_w [k * 8 + c];
  else if (c < 16) v = te1_w[k * 8 + (c - 8)];
  else if (c < 24) v = te2_w[k * 8 + (c - 16)];
  else             v = te3_w[k * 8 + (c - 24)];
  ws[WS_WCAT + e] = v;
}

// ------------------------------------------------- K1: per-node projections
__global__ __launch_bounds__(128) void prep_kernel(
    const float* __restrict__ node, const float* __restrict__ hidden,
    const float* __restrict__ graph,
    const float* __restrict__ m_w, const float* __restrict__ a1_w,
    const float* __restrict__ a2_w, const float* __restrict__ skip_w,
    const float* __restrict__ skip_b,
    const float* __restrict__ t1_w, const float* __restrict__ t1_b,
    const float* __restrict__ t2_w, const float* __restrict__ t2_b,
    const float* __restrict__ ag_w, const float* __restrict__ tg_w,
    const float* __restrict__ tg_b, float* __restrict__ ws) {
  __shared__ float z[ZD];
  int bn = blockIdx.x;              // b*N + n
  int tid = threadIdx.x;            // 0..127 -> output channel
  z[tid]         = node  [bn * INdim + tid];
  z[INdim + tid] = hidden[bn * INdim + tid];
  __syncthreads();

  float av = 0.f, as = skip_b[tid];
  for (int f = 0; f < ZD; ++f) {
    float zf = z[f];
    av += zf * m_w   [f * OUTdim + tid];
    as += zf * skip_w[f * OUTdim + tid];
  }
  ws[WS_VALUES + bn * OUTdim + tid] = av;
  ws[WS_SKIP   + bn * OUTdim + tid] = as;

  if (tid < Hdim) {
    float s1 = 0.f, s2 = 0.f;
    for (int f = 0; f < ZD; ++f) {
      float zf = z[f];
      s1 += zf * a1_w[f * Hdim + tid];
      s2 += zf * a2_w[f * Hdim + tid];
    }
    ws[WS_A1 + bn * Hdim + tid] = s1;
    ws[WS_A2 + bn * Hdim + tid] = s2;
  } else if (tid < 16) {
    int t = tid - 8;
    float s1 = t1_b[t], s2 = t2_b[t];
    for (int f = 0; f < ZD; ++f) {
      float zf = z[f];
      s1 += zf * t1_w[f * Tdim + t];
      s2 += zf * t2_w[f * Tdim + t];
    }
    ws[WS_TRI1 + bn * Tdim + t] = s1;
    ws[WS_TRI2 + bn * Tdim + t] = s2;
  }
  int b = bn >> 7, n = bn & 127;
  if (n == 0 && tid < Hdim) {
    float sa = 0.f, sg = tg_b[tid];
    for (int f = 0; f < INdim; ++f) {
      float gf = graph[b * INdim + f];
      sa += gf * ag_w[f * Hdim + tid];
      sg += gf * tg_w[f * Tdim + tid];
    }
    ws[WS_AG + b * Hdim + tid] = sa;
    ws[WS_G  + b * Tdim + tid] = sg;
  }
}

// -------------------------------- K2: (65536x128)@(128x32) via WMMA f32 16x16x4
__global__ __launch_bounds__(256) void edge_gemm_kernel(
    const float* __restrict__ edge,
    const float* __restrict__ te1_b, const float* __restrict__ te2_b,
    const float* __restrict__ te3_b, float* __restrict__ ws) {
  __shared__ float Es[128 * 128];         // 64 KB A-tile
  const float* Wc = ws + WS_WCAT;
  int m_base = blockIdx.x * 128;          // 512 blocks
  int tid = threadIdx.x;

  // cooperative coalesced stage of 128 rows x 128 floats
  const float4* src = (const float4*)(edge + (size_t)m_base * INdim);
  float4* dst = (float4*)Es;
  #pragma unroll
  for (int q = 0; q < 16; ++q) dst[q * 256 + tid] = src[q * 256 + tid];
  __syncthreads();

  int wv = tid >> 5, lane = tid & 31;
  int lrow = lane & 15, khalf = lane >> 4;
  int row_l = wv * 16 + lrow;             // A row (M) for this lane
  v8f c0 = {}; v8f c1 = {};
  for (int k0 = 0; k0 < 128; k0 += 4) {
    int kk = k0 + 2 * khalf;
    v2f a;                                // A: lane=M, vgpr0: K=kk, vgpr1: K=kk+1
    a.x = Es[row_l * 128 + kk];
    a.y = Es[row_l * 128 + kk + 1];
    v2f b0, b1;                           // B: lane=N, same K striping as A
    b0.x = Wc[kk * EC + lrow];       b0.y = Wc[(kk + 1) * EC + lrow];
    b1.x = Wc[kk * EC + 16 + lrow];  b1.y = Wc[(kk + 1) * EC + 16 + lrow];
    c0 = __builtin_amdgcn_wmma_f32_16x16x4_f32(false, a, false, b0, (short)0, c0, false, false);
    c1 = __builtin_amdgcn_wmma_f32_16x16x4_f32(false, a, false, b1, (short)0, c1, false, false);
  }
  // biases: cols 0-7 none (ae), 8-15 te1_b, 16-23 te2_b, 24-31 te3_b
  int col0 = lrow, col1 = 16 + lrow;
  float bias0 = (col0 < 8) ? 0.f : te1_b[col0 - 8];
  float bias1 = (col1 < 24) ? te2_b[col1 - 16] : te3_b[col1 - 24];
  float* O = ws + WS_EDGE;
  #pragma unroll
  for (int v = 0; v < 8; ++v) {           // C: lane=N, row M = v + 8*khalf
    size_t R = (size_t)m_base + wv * 16 + v + 8 * khalf;
    O[R * EC + col0] = c0[v] + bias0;
    O[R * EC + col1] = c1[v] + bias1;
  }
}

// --------------------------------- K3: attention softmax + aggregate + LN
__global__ __launch_bounds__(128) void attn_kernel(
    const float* __restrict__ adj, const float* __restrict__ gamma,
    const float* __restrict__ beta, const float* __restrict__ ws,
    float* __restrict__ out) {
  __shared__ float lg[Hdim * Ndim];   // logits -> coefs in place
  __shared__ float red[128];
  int bi = blockIdx.x; int b = bi >> 7;
  int tid = threadIdx.x;

  for (int idx = tid; idx < Hdim * Ndim; idx += 128) {
    int h = idx >> 7, j = idx & 127;
    float v = ws[WS_A1 + bi * Hdim + h]
            + ws[WS_A2 + (b * Ndim + j) * Hdim + h]
            + ws[WS_EDGE + ((size_t)bi * Ndim + j) * EC + h]
            + ws[WS_AG + b * Hdim + h];
    v = (v > 0.f) ? v : 0.01f * v;        // leaky_relu(0.01)
    lg[idx] = v;
  }
  __syncthreads();
  if (tid < Hdim) {
    int h = tid;
    float m = -INFINITY;
    for (int j = 0; j < Ndim; ++j) m = fmaxf(m, lg[h * Ndim + j]);
    float s = 0.f;
    for (int j = 0; j < Ndim; ++j) s += expf(lg[h * Ndim + j] - m);
    float inv = 1.f / s;
    for (int j = 0; j < Ndim; ++j)
      lg[h * Ndim + j] = adj[(size_t)bi * Ndim + j] * expf(lg[h * Ndim + j] - m) * inv;
  }
  __syncthreads();
  int h = tid >> 4;                       // head of this output channel
  float acc = 0.f;
  for (int j = 0; j < Ndim; ++j)
    acc += lg[h * Ndim + j] * ws[WS_VALUES + ((size_t)b * Ndim + j) * OUTdim + tid];
  acc += ws[WS_SKIP + (size_t)bi * OUTdim + tid];
  acc = fmaxf(acc, 0.f);
  // layernorm over 128 channels
  red[tid] = acc; __syncthreads();
  for (int s = 64; s; s >>= 1) { if (tid < s) red[tid] += red[tid + s]; __syncthreads(); }
  float mu = red[0] * (1.f / OUTdim); __syncthreads();
  float d = acc - mu;
  red[tid] = d * d; __syncthreads();
  for (int s = 64; s; s >>= 1) { if (tid < s) red[tid] += red[tid + s]; __syncthreads(); }
  float var = red[0] * (1.f / OUTdim);
  out[(size_t)bi * OUTdim + tid] = d * rsqrtf(var + EPSc) * gamma[tid] + beta[tid];
}

// ------------------- K4: triplet max-reduction + projection + relu + LN
__global__ __launch_bounds__(256) void tri_kernel(
    const float* __restrict__ o3_w, const float* __restrict__ o3_b,
    const float* __restrict__ gamma, const float* __restrict__ beta,
    const float* __restrict__ ws, float* __restrict__ out) {
  __shared__ float e1s[32 * 16 * 8];    // [ii][jj][t], tri1 folded in (16 KB)
  __shared__ float e2s[32 * 16 * 8];    // [ii][kk][t] (16 KB)
  __shared__ float tris[256 * 8];       // per-pair triplet vector (8 KB)
  __shared__ float2 wred[8];
  int b = blockIdx.z;
  int J0 = blockIdx.x * 16, K0 = blockIdx.y * 16;
  int tid = threadIdx.x;
  int jj = tid >> 4, kk = tid & 15;     // this thread's (j,k) pair in tile
  const float* edge = ws + WS_EDGE;

  float m0[8];
  #pragma unroll
  for (int t = 0; t < 8; ++t) m0[t] = -INFINITY;

  for (int c = 0; c < 4; ++c) {         // i in chunks of 32
    int i0 = c * 32;
    int ii = tid >> 3;
    int j2 = (tid & 7) * 2;
    #pragma unroll
    for (int s = 0; s < 2; ++s) {
      int jc = j2 + s;
      const float* er1 = edge + (((size_t)(b * Ndim + i0 + ii) * Ndim) + (J0 + jc)) * EC + 8;
      const float* tr1 = ws + WS_TRI1 + (b * Ndim + i0 + ii) * Tdim;
      float* d1 = e1s + (ii * 16 + jc) * 8;
      #pragma unroll
      for (int t = 0; t < 8; ++t) d1[t] = er1[t] + tr1[t];
      const float* er2 = edge + (((size_t)(b * Ndim + i0 + ii) * Ndim) + (K0 + jc)) * EC + 16;
      float* d2 = e2s + (ii * 16 + jc) * 8;
      #pragma unroll
      for (int t = 0; t < 8; ++t) d2[t] = er2[t];
    }
    __syncthreads();
    for (int ii2 = 0; ii2 < 32; ++ii2) {
      const float* p1 = e1s + (ii2 * 16 + jj) * 8;
      const float* p2 = e2s + (ii2 * 16 + kk) * 8;
      #pragma unroll
      for (int t = 0; t < 8; ++t) m0[t] = fmaxf(m0[t], p1[t] + p2[t]);
    }
    __syncthreads();
  }

  {
    int j = J0 + jj, k = K0 + kk;
    const float* t2 = ws + WS_TRI2 + (b * Ndim + j) * Tdim;
    const float* t3 = ws + WS_TRI1 + (b * Ndim + k) * Tdim;   // tri3 reuses t1_w
    const float* e3 = edge + (((size_t)(b * Ndim + j) * Ndim) + k) * EC + 24;
    const float* g  = ws + WS_G + b * Tdim;
    #pragma unroll
    for (int t = 0; t < 8; ++t)
      tris[tid * 8 + t] = m0[t] + t2[t] + t3[t] + e3[t] + g[t];
  }
  __syncthreads();

  // projection (8 -> 128) + relu + rowwise LN; two 128-thread groups, coalesced
  int group = tid >> 7;
  int o = tid & 127;
  int wv = tid >> 5;
  for (int it = 0; it < 128; ++it) {
    int p = it * 2 + group;
    const float* tp = tris + p * 8;
    float v = o3_b[o];
    #pragma unroll
    for (int t = 0; t < 8; ++t) v += tp[t] * o3_w[t * OUTdim + o];
    v = fmaxf(v, 0.f);
    float s = v, s2 = v * v;
    #pragma unroll
    for (int off = 16; off; off >>= 1) {
      s  += __shfl_xor(s, off);
      s2 += __shfl_xor(s2, off);
    }
    if ((tid & 31) == 0) wred[wv] = make_float2(s, s2);
    __syncthreads();
    int base = group * 4;
    float fs  = wred[base].x + wred[base + 1].x + wred[base + 2].x + wred[base + 3].x;
    float fs2 = wred[base].y + wred[base + 1].y + wred[base + 2].y + wred[base + 3].y;
    float mu  = fs * (1.f / OUTdim);
    float var = fs2 * (1.f / OUTdim) - mu * mu;
    int pj = p >> 4, pk = p & 15;
    size_t row = ((size_t)(b * Ndim + J0 + pj) * Ndim) + (K0 + pk);
    out[row * OUTdim + o] = (v - mu) * rsqrtf(var + EPSc) * gamma[o] + beta[o];
    __syncthreads();
  }
}

// ----------------------------------------------------------------- launcher
extern "C" void kernel_launch(void* const* d_in, const int* in_sizes, int n_in,
                              void* d_out, int out_size, void* d_ws, size_t ws_size,
                              hipStream_t stream) {
  const float* node   = (const float*)d_in[0];
  const float* edge   = (const float*)d_in[1];
  const float* graph  = (const float*)d_in[2];
  const float* hidden = (const float*)d_in[3];
  const float* adj    = (const float*)d_in[4];
  const float* m_w    = (const float*)d_in[5];
  const float* a1_w   = (const float*)d_in[6];
  const float* a2_w   = (const float*)d_in[7];
  const float* ae_w   = (const float*)d_in[8];
  const float* ag_w   = (const float*)d_in[9];
  const float* skip_w = (const float*)d_in[10];
  const float* skip_b = (const float*)d_in[11];
  const float* gamma  = (const float*)d_in[12];
  const float* beta   = (const float*)d_in[13];
  const float* t1_w   = (const float*)d_in[14];
  const float* t1_b   = (const float*)d_in[15];
  const float* t2_w   = (const float*)d_in[16];
  const float* t2_b   = (const float*)d_in[17];
  const float* te1_w  = (const float*)d_in[18];
  const float* te1_b  = (const float*)d_in[19];
  const float* te2_w  = (const float*)d_in[20];
  const float* te2_b  = (const float*)d_in[21];
  const float* te3_w  = (const float*)d_in[22];
  const float* te3_b  = (const float*)d_in[23];
  const float* tg_w   = (const float*)d_in[24];
  const float* tg_b   = (const float*)d_in[25];
  const float* o3_w   = (const float*)d_in[26];
  const float* o3_b   = (const float*)d_in[27];
  float* ws  = (float*)d_ws;
  float* out = (float*)d_out;

  build_wcat_kernel<<<16, 256, 0, stream>>>(ae_w, te1_w, te2_w, te3_w, ws);
  prep_kernel<<<Bdim * Ndim, 128, 0, stream>>>(node, hidden, graph, m_w, a1_w,
      a2_w, skip_w, skip_b, t1_w, t1_b, t2_w, t2_b, ag_w, tg_w, tg_b, ws);
  edge_gemm_kernel<<<(Bdim * Ndim * Ndim) / 128, 256, 0, stream>>>(
      edge, te1_b, te2_b, te3_b, ws);
  attn_kernel<<<Bdim * Ndim, 128, 0, stream>>>(adj, gamma, beta, ws, out);
  tri_kernel<<<dim3(Ndim / 16, Ndim / 16, Bdim), 256, 0, stream>>>(
      o3_w, o3_b, gamma, beta, ws, out + Bdim * Ndim * OUTdim);
}